// GATv2_59184649339075
// MI455X (gfx1250) — compile-verified
//
#include <hip/hip_runtime.h>
#include <hip/hip_bf16.h>
#include <math.h>

#define DD 64
#define NSLOPE 0.2f

typedef __attribute__((ext_vector_type(2))) float v2f;
typedef __attribute__((ext_vector_type(8))) float v8f;

// Order-preserving float -> uint mapping for atomicMax-based segment max.
__device__ __forceinline__ unsigned flipF(float f) {
    unsigned u = __float_as_uint(f);
    return (u >> 31) ? ~u : (u | 0x80000000u);
}
__device__ __forceinline__ float unflipF(unsigned u) {
    unsigned v = (u >> 31) ? (u & 0x7FFFFFFFu) : ~u;
    return __uint_as_float(v);
}

// ---------------------------------------------------------------------------
// Kernel 0: zero agg (= d_out), denom, emax
// ---------------------------------------------------------------------------
__global__ void k_init(float* __restrict__ agg, float* __restrict__ denom,
                       unsigned* __restrict__ emaxU, int n) {
    int i = blockIdx.x * blockDim.x + threadIdx.x;
    if (i < n * DD) agg[i] = 0.0f;
    if (i < n) { denom[i] = 0.0f; emaxU[i] = 0u; }  // 0 == flip(-NaN), below any real
}

// ---------------------------------------------------------------------------
// Kernel 1: Wh = x @ W^T + b via v_wmma_f32_16x16x4_f32.
// One wave per 16-node tile; A fragments (x rows) hoisted across all K chunks.
// A layout (16x4 f32): lane<16 -> M=lane, {K0,K0+1}; lane>=16 -> M=lane-16, {K0+2,K0+3}
// B layout (4x16 f32): mirrored: lane half selects K row pair, lm selects column N
// D layout: VGPR v -> row v + 8*half, col = lm
// ---------------------------------------------------------------------------
__global__ __launch_bounds__(256) void k_gemm(const float* __restrict__ x,
                                              const float* __restrict__ W,
                                              const float* __restrict__ bias,
                                              float* __restrict__ Wh, int n) {
    const int lane = threadIdx.x & 31;
    const int wave = threadIdx.x >> 5;
    const int tile = blockIdx.x * 8 + wave;
    if (tile * 16 >= n) return;          // uniform across the wave (EXEC all-1 for WMMA)

    const int lm   = lane & 15;
    const int half = lane >> 4;
    const int row  = tile * 16 + lm;

    // Hoist all A fragments: lane covers x[row][4*i + 2*half + {0,1}], i=0..15
    const float* xr = x + (size_t)row * DD + 2 * half;
    v2f areg[16];
#pragma unroll
    for (int i = 0; i < 16; ++i) {
        areg[i].x = xr[4 * i + 0];
        areg[i].y = xr[4 * i + 1];
    }

#pragma unroll
    for (int c = 0; c < 4; ++c) {        // 4 column tiles of 16 output features
        v8f acc = {};
        const float* wr = W + (size_t)(c * 16 + lm) * DD + 2 * half;  // B = W^T tile
#pragma unroll
        for (int i = 0; i < 16; ++i) {   // K = 64 in chunks of 4
            v2f bfrag;
            bfrag.x = wr[4 * i + 0];
            bfrag.y = wr[4 * i + 1];
            acc = __builtin_amdgcn_wmma_f32_16x16x4_f32(
                false, areg[i], false, bfrag, (short)0, acc, false, false);
        }
        const float bv = bias[c * 16 + lm];
#pragma unroll
        for (int v = 0; v < 8; ++v) {
            Wh[(size_t)(tile * 16 + v + 8 * half) * DD + c * 16 + lm] = acc[v] + bv;
        }
    }
}

// ---------------------------------------------------------------------------
// Kernel 2: per-edge attention logits e = a . leakyrelu(Wh[dst]+Wh[src]),
// plus segment max via flipped-uint atomicMax. One wave32 per edge.
// ---------------------------------------------------------------------------
__global__ __launch_bounds__(256) void k_logits(const float* __restrict__ Wh,
                                                const float* __restrict__ avec,
                                                const int* __restrict__ src,
                                                const int* __restrict__ dst,
                                                float* __restrict__ eb,
                                                unsigned* __restrict__ emaxU, int E) {
    const int lane = threadIdx.x & 31;
    const int e = blockIdx.x * 8 + (threadIdx.x >> 5);
    if (e >= E) return;
    const int s = src[e], d = dst[e];

    const float2 ws = ((const float2*)(Wh + (size_t)s * DD))[lane];
    const float2 wd = ((const float2*)(Wh + (size_t)d * DD))[lane];
    const float2 av = ((const float2*)avec)[lane];

    float sx = ws.x + wd.x, sy = ws.y + wd.y;
    sx = sx > 0.0f ? sx : NSLOPE * sx;
    sy = sy > 0.0f ? sy : NSLOPE * sy;
    float p = sx * av.x + sy * av.y;

#pragma unroll
    for (int m = 16; m >= 1; m >>= 1) p += __shfl_xor(p, m, 32);   // wave32 reduce

    if (lane == 0) {
        eb[e] = p;
        atomicMax(emaxU + d, flipF(p));
    }
}

// ---------------------------------------------------------------------------
// Kernel 3: ex = exp(e - emax[dst]) (in place), denom[dst] += ex
// ---------------------------------------------------------------------------
__global__ void k_exp(const int* __restrict__ dst, float* __restrict__ eb,
                      const unsigned* __restrict__ emaxU,
                      float* __restrict__ denom, int E) {
    int i = blockIdx.x * blockDim.x + threadIdx.x;
    if (i >= E) return;
    int d = dst[i];
    float ex = __expf(eb[i] - unflipF(emaxU[d]));
    eb[i] = ex;
    unsafeAtomicAdd(denom + d, ex);
}

// ---------------------------------------------------------------------------
// Kernel 4: agg[dst] += alpha * Wh[src]; one wave per edge, 2 dims per lane.
// fp32 HW atomics hit L2 (agg is 12.8MB, L2-resident on 192MB L2).
// ---------------------------------------------------------------------------
__global__ __launch_bounds__(256) void k_agg(const float* __restrict__ Wh,
                                             const int* __restrict__ src,
                                             const int* __restrict__ dst,
                                             const float* __restrict__ eb,
                                             const float* __restrict__ denom,
                                             float* __restrict__ agg, int E) {
    const int lane = threadIdx.x & 31;
    const int e = blockIdx.x * 8 + (threadIdx.x >> 5);
    if (e >= E) return;
    const int s = src[e], d = dst[e];
    const float alpha = eb[e] / fmaxf(denom[d], 1e-9f);
    const float2 w = ((const float2*)(Wh + (size_t)s * DD))[lane];
    float* ap = agg + (size_t)d * DD + 2 * lane;
    unsafeAtomicAdd(ap + 0, alpha * w.x);
    unsafeAtomicAdd(ap + 1, alpha * w.y);
}

// ---------------------------------------------------------------------------
// Kernel 5: out = sigmoid(agg) in place (empty nodes: sigmoid(0)=0.5, matches ref)
// ---------------------------------------------------------------------------
__global__ void k_sig(float* __restrict__ out, int n) {
    int i = blockIdx.x * blockDim.x + threadIdx.x;
    if (i < n * DD) out[i] = 1.0f / (1.0f + __expf(-out[i]));
}

// ---------------------------------------------------------------------------
extern "C" void kernel_launch(void* const* d_in, const int* in_sizes, int n_in,
                              void* d_out, int out_size, void* d_ws, size_t ws_size,
                              hipStream_t stream) {
    const float* x    = (const float*)d_in[0];   // [N, 64]
    const float* W    = (const float*)d_in[1];   // [64, 64]
    const float* bias = (const float*)d_in[2];   // [64]
    const float* avec = (const float*)d_in[3];   // [64]
    const int*   ei   = (const int*)d_in[4];     // [2, E]

    const int N = in_sizes[0] / DD;
    const int E = in_sizes[4] / 2;
    const int* src = ei;
    const int* dst = ei + E;

    // Workspace: Wh [N*64], eb [E], emaxU [N], denom [N]  (~16.4 MB)
    float*    Wh    = (float*)d_ws;
    float*    eb    = Wh + (size_t)N * DD;
    unsigned* emaxU = (unsigned*)(eb + E);
    float*    denom = (float*)(emaxU + N);

    float* agg = (float*)d_out;   // accumulate directly into output buffer

    const int nd = N * DD;
    k_init<<<(nd + 255) / 256, 256, 0, stream>>>(agg, denom, emaxU, N);

    const int tiles = (N + 15) / 16;
    k_gemm<<<(tiles + 7) / 8, 256, 0, stream>>>(x, W, bias, Wh, N);

    k_logits<<<(E + 7) / 8, 256, 0, stream>>>(Wh, avec, src, dst, eb, emaxU, E);

    k_exp<<<(E + 255) / 256, 256, 0, stream>>>(dst, eb, emaxU, denom, E);

    k_agg<<<(E + 7) / 8, 256, 0, stream>>>(Wh, src, dst, eb, denom, agg, E);

    k_sig<<<(nd + 255) / 256, 256, 0, stream>>>(agg, N);
}